// DepthwiseCorr_89472758710581
// MI455X (gfx1250) — compile-verified
//
#include <hip/hip_runtime.h>
#include <stdint.h>

// Depthwise cross-correlation, f32, NHWC.
// search:   [64, 31, 31, 256]
// template: [64,  7,  7, 256]
// out:      [64, 25, 25, 256]
//
// Roofline: ~107 MB traffic @ 23.3 TB/s => ~4.6 us floor; 1.0 GFLOP f32.
// Depthwise => no channel reduction => WMMA cannot be engaged without >=16x
// waste (per-channel matvec, N=1). Optimal CDNA5 path: TDM tensor DMA of the
// strided channel-group tile into LDS + dense f32 VALU FMA with full reuse.

#define CH 256
#define XS 31          // search spatial
#define KS 7           // kernel spatial
#define OS 25          // output spatial
#define CG 32          // channels per group == wave32 lanes
#define NG (CH / CG)   // 8 channel groups
#define NSPLIT 3       // row splits per (batch, group)
#define MAXROWS 9      // max output rows per workgroup
#define MAXSROWS (MAXROWS + KS - 1)   // 15 search rows
#define BLOCK (MAXROWS * 32)          // 288 threads = 9 waves
#define TILE_BYTES (MAXSROWS * XS * CG * 4)  // 59520 B LDS

typedef unsigned int u32x4 __attribute__((ext_vector_type(4)));
typedef int i32x8 __attribute__((ext_vector_type(8)));
typedef int i32x4 __attribute__((ext_vector_type(4)));

#if __has_builtin(__builtin_amdgcn_tensor_load_to_lds)
// One TDM descriptor: 2D tile, inner dim = rowlen elems (contiguous 4B),
// nrows rows at 'stride' elems between row starts. Dest LDS is fully packed.
__device__ __forceinline__ void tdm_load_rows_to_lds(const float* gsrc,
                                                     unsigned lds_byte_off,
                                                     unsigned nrows,
                                                     unsigned rowlen,
                                                     unsigned stride_elems) {
  unsigned long long ga = (unsigned long long)(uintptr_t)gsrc;
  u32x4 g0;
  g0[0] = 1u;                                   // count=1, is_restore=0, no gather
  g0[1] = lds_byte_off;                         // lds_addr (bytes)
  g0[2] = (unsigned)ga;                         // global_addr[31:0]  (bits 95:64)
  g0[3] = (unsigned)((ga >> 32) & 0x01FFFFFFu)  // global_addr[56:32] (bits 120:96)
          | (2u << 30);                         // type=2 ("image")
  i32x8 g1;
  g1[0] = (int)(2u << 16);                      // wg_mask=0, data_size=2 (4 bytes)
  g1[1] = (int)((rowlen & 0xFFFFu) << 16);      // tensor_dim0[15:0] @ bits 63:48
  g1[2] = (int)(((rowlen >> 16) & 0xFFFFu)      // tensor_dim0[31:16]
          | ((nrows & 0xFFFFu) << 16));         // tensor_dim1[15:0]
  g1[3] = (int)(((nrows >> 16) & 0xFFFFu)       // tensor_dim1[31:16]
          | ((rowlen & 0xFFFFu) << 16));        // tile_dim0
  g1[4] = (int)(nrows & 0xFFFFu);               // tile_dim1 ; tile_dim2 = 0 (2D)
  g1[5] = (int)stride_elems;                    // tensor_dim0_stride[31:0]
  g1[6] = 0;                                    // stride hi / dim1_stride lo
  g1[7] = 0;
  i32x4 g2 = (i32x4)0;
  i32x4 g3 = (i32x4)0;
#if __clang_major__ >= 23
  i32x8 g4 = (i32x8)0;
  __builtin_amdgcn_tensor_load_to_lds(g0, g1, g2, g3, g4, 0);
#else
  __builtin_amdgcn_tensor_load_to_lds(g0, g1, g2, g3, 0);
#endif
}
#endif

__global__ __launch_bounds__(BLOCK) void dwcorr_tdm_kernel(
    const float* __restrict__ search,
    const float* __restrict__ tmpl,
    float* __restrict__ out) {
  const int blk = blockIdx.x;
  const int b   = blk / (NG * NSPLIT);
  const int rem = blk % (NG * NSPLIT);
  const int g   = rem / NSPLIT;
  const int r   = rem % NSPLIT;
  const int rowStart = (r == 0) ? 0 : ((r == 1) ? 9 : 17);
  const int nRows    = (r == 0) ? 9 : 8;        // output rows in this split
  const int sRows    = nRows + KS - 1;          // search rows needed

  const int lane = threadIdx.x & 31;            // channel within group (wave32)
  const int wave = threadIdx.x >> 5;            // output row within split

  extern __shared__ float tile[];               // [sRows*31][32] packed f32

  // Global base of staged region: search[b][rowStart][0][g*CG].
  // (y,x) pixels flatten to a single 2D walk: 128B rows at 1KB (256-elem) stride.
  const float* gsrc =
      search + (((size_t)b * XS + rowStart) * XS) * CH + (size_t)g * CG;

  // Template taps -> registers (49 coalesced 128B loads, overlaps the DMA).
  float tk[KS][KS];
  {
    const float* tb = tmpl + ((size_t)b * KS * KS) * CH + (size_t)g * CG + lane;
    #pragma unroll
    for (int i = 0; i < KS; ++i)
      #pragma unroll
      for (int j = 0; j < KS; ++j)
        tk[i][j] = tb[(size_t)(i * KS + j) * CH];
  }

#if __has_builtin(__builtin_amdgcn_tensor_load_to_lds)
  if (threadIdx.x < 32) {                       // wave 0 only: TDM is per-wave DMA
    const unsigned lds_off = (unsigned)(uintptr_t)tile;  // low 32b = LDS offset
    tdm_load_rows_to_lds(gsrc, lds_off, (unsigned)(sRows * XS), CG, CH);
    __builtin_amdgcn_s_wait_tensorcnt(0);
  }
#else
#warning "tensor_load_to_lds builtin unavailable: using cooperative-copy fallback"
  for (int idx = threadIdx.x; idx < sRows * XS * CG; idx += BLOCK) {
    const int row = idx / CG;
    const int c   = idx % CG;
    tile[idx] = gsrc[(size_t)row * CH + c];
  }
#endif
  __syncthreads();

  if (wave < nRows) {
    const int h = rowStart + wave;              // global output row

    float acc[OS];
    #pragma unroll
    for (int w = 0; w < OS; ++w) acc[w] = 0.0f;

    #pragma unroll
    for (int i = 0; i < KS; ++i) {
      // Search row (wave+i) of the tile: per-lane stride CG floats, conflict-free.
      float srow[XS];
      const int rb = ((wave + i) * XS) * CG + lane;
      #pragma unroll
      for (int x = 0; x < XS; ++x) srow[x] = tile[rb + x * CG];

      #pragma unroll
      for (int j = 0; j < KS; ++j) {
        const float tv = tk[i][j];
        #pragma unroll
        for (int w = 0; w < OS; ++w)
          acc[w] = __builtin_fmaf(srow[w + j], tv, acc[w]);
      }
    }

    float* ob = out + (((size_t)b * OS + h) * OS) * CH + (size_t)g * CG + lane;
    #pragma unroll
    for (int w = 0; w < OS; ++w) ob[(size_t)w * CH] = acc[w];
  }
}

extern "C" void kernel_launch(void* const* d_in, const int* in_sizes, int n_in,
                              void* d_out, int out_size, void* d_ws, size_t ws_size,
                              hipStream_t stream) {
  (void)in_sizes; (void)n_in; (void)out_size; (void)d_ws; (void)ws_size;
  const float* search = (const float*)d_in[0];
  const float* tmpl   = (const float*)d_in[1];
  float* out          = (float*)d_out;

  dim3 grid(64 * NG * NSPLIT);   // 1536 workgroups
  dim3 block(BLOCK);             // 288 threads = 9 waves
  dwcorr_tdm_kernel<<<grid, block, TILE_BYTES, stream>>>(search, tmpl, out);
}